// MultiResHashGridMLP_31550829756944
// MI455X (gfx1250) — compile-verified
//
#include <hip/hip_runtime.h>
#include <hip/hip_bf16.h>
#include <stdint.h>

// ---------------------------------------------------------------------------
// Instant-NGP multires hash-grid encode, gfx1250 (wave32).
// 1M points x 16 levels x 8-corner float2 gathers. All 16 tables = 64MB,
// fully resident in MI455X's 192MB L2 -> kernel is L2-gather bound.
// Key decisions:
//   * 8 independent global_load_b64 gathers in flight per thread per level
//     (full unroll), register-light for high wave32 occupancy.
//   * Streaming traffic (x in: 12MB, out: 140MB) uses NT temporal hints so
//     it does NOT evict the L2-resident tables.
//   * Output coalesced via per-wave LDS staging: each store instruction
//     writes one full 128B cacheline (32 lanes x b32 contiguous).
//   * Non-power-of-2 hash moduli are compile-time constants -> mul-hi magic,
//     no integer division.
// ---------------------------------------------------------------------------

#define NLEVELS   16
#define TBL_N     (1u << 19)          // entries per level
#define TBL_MASK  (TBL_N - 1u)
#define OUT_COLS  35                  // 3 coords + 16*2 feats
#define P1        2654435761u
#define P2        805459861u

// res[l] = floor(16 * 128^(l/15));  hs[l] = min(res^3, 2^19)
__device__ constexpr float    kRes[NLEVELS] = {
    16.f, 22.f, 30.f, 42.f, 58.f, 80.f, 111.f, 153.f,
    212.f, 294.f, 406.f, 561.f, 776.f, 1072.f, 1481.f, 2048.f};
__device__ constexpr uint32_t kHS[NLEVELS] = {
    4096u, 10648u, 27000u, 74088u, 195112u, 512000u,
    TBL_N, TBL_N, TBL_N, TBL_N, TBL_N, TBL_N, TBL_N, TBL_N, TBL_N, TBL_N};

__global__ __launch_bounds__(256)
void hashgrid_encode_kernel(const float* __restrict__ x,
                            const float* __restrict__ tables,
                            float* __restrict__ out,
                            int n)
{
    // Per-wave staging: 8 waves * 32 lanes * 35 floats = 35,840 B LDS
    // (trivial vs the 320KB WGP pool; no occupancy impact).
    __shared__ float stage[8 * 32 * OUT_COLS];

    const int lane = threadIdx.x & 31;     // gfx1250 is wave32
    const int wv   = threadIdx.x >> 5;
    const int i    = blockIdx.x * blockDim.x + threadIdx.x;
    float* sw = &stage[wv * (32 * OUT_COLS)];

    // Streamed once -> non-temporal: don't let x claim L2 lines the
    // gather tables need.
    float px = 0.f, py = 0.f, pz = 0.f;
    if (i < n) {
        px = __builtin_nontemporal_load(x + 3 * i + 0);
        py = __builtin_nontemporal_load(x + 3 * i + 1);
        pz = __builtin_nontemporal_load(x + 3 * i + 2);
    }
    // Passthrough coords into staging row (stride 35 words: gcd(35,64)=1 ->
    // bank-conflict-free column writes).
    sw[lane * OUT_COLS + 0] = px;
    sw[lane * OUT_COLS + 1] = py;
    sw[lane * OUT_COLS + 2] = pz;

    const float2* __restrict__ tab2 = (const float2*)tables;

#pragma unroll
    for (int l = 0; l < NLEVELS; ++l) {
        const float r = kRes[l];
        const float xs0 = px * r, xs1 = py * r, xs2 = pz * r;
        const int   i0 = (int)xs0, i1 = (int)xs1, i2 = (int)xs2;
        const float f0 = xs0 - (float)i0;
        const float f1 = xs1 - (float)i1;
        const float f2 = xs2 - (float)i2;

        // Per-axis hash contributions for corner bit {0,1}.
        const uint32_t A[2] = { (uint32_t)i0,        (uint32_t)i0 + 1u };
        const uint32_t B[2] = { (uint32_t)i1 * P1,   (uint32_t)i1 * P1 + P1 };
        const uint32_t C[2] = { (uint32_t)i2 * P2,   (uint32_t)i2 * P2 + P2 };
        const float   WA[2] = { 1.f - f0, f0 };
        const float   WB[2] = { 1.f - f1, f1 };
        const float   WC[2] = { 1.f - f2, f2 };

        const float2* __restrict__ tl = tab2 + (size_t)l * TBL_N;

        // Phase 1: compute all 8 hashed indices, issue all 8 b64 gathers
        // back-to-back (8 outstanding L2 loads per thread). Default RT
        // temporal policy: these are the lines we WANT resident in L2.
        float2 t[8];
        float  w[8];
#pragma unroll
        for (int c = 0; c < 8; ++c) {
            const int b0 = c & 1, b1 = (c >> 1) & 1, b2 = (c >> 2) & 1;
            const uint32_t h = A[b0] ^ B[b1] ^ C[b2];
            // kHS[l] is a compile-time constant after full unroll:
            // power-of-2 levels -> AND; small levels -> mul-hi magic.
            const uint32_t hid = (kHS[l] == TBL_N) ? (h & TBL_MASK)
                                                   : (h % kHS[l]);
            t[c] = tl[hid];
            w[c] = WA[b0] * WB[b1] * WC[b2];
        }

        // Phase 2: trilinear blend.
        float a0 = 0.f, a1 = 0.f;
#pragma unroll
        for (int c = 0; c < 8; ++c) {
            a0 = fmaf(w[c], t[c].x, a0);
            a1 = fmaf(w[c], t[c].y, a1);
        }

        sw[lane * OUT_COLS + 3 + 2 * l + 0] = a0;
        sw[lane * OUT_COLS + 3 + 2 * l + 1] = a1;
    }

    // Waves only touch their own LDS region, but sync for cross-lane LDS
    // visibility before the coalesced copy-out.
    __syncthreads();

    // Coalesced write-out: this wave's 32 rows are 1120 contiguous floats in
    // both LDS and global memory. Each iteration = one full 128B cacheline
    // per wave. NT store: 140MB output stream must not evict the tables.
    const long long outBase =
        ((long long)blockIdx.x * blockDim.x + (long long)wv * 32) * OUT_COLS;
    const long long outEnd = (long long)n * OUT_COLS;
#pragma unroll 1
    for (int k = lane; k < 32 * OUT_COLS; k += 32) {
        const long long g = outBase + k;
        if (g < outEnd) __builtin_nontemporal_store(sw[k], out + g);
    }
}

extern "C" void kernel_launch(void* const* d_in, const int* in_sizes, int n_in,
                              void* d_out, int out_size, void* d_ws, size_t ws_size,
                              hipStream_t stream) {
    const float* x      = (const float*)d_in[0];   // (N, 3) f32
    const float* tables = (const float*)d_in[1];   // (16, 2^19, 2) f32
    float*       out    = (float*)d_out;           // (N, 35) f32

    const int n = in_sizes[0] / 3;
    const int block = 256;
    const int grid  = (n + block - 1) / block;
    hashgrid_encode_kernel<<<grid, block, 0, stream>>>(x, tables, out, n);
}